// GlobalPool_21723944583658
// MI455X (gfx1250) — compile-verified
//
#include <hip/hip_runtime.h>

// CDNA5 / gfx1250: wave32, V_WMMA_F32_16X16X4_F32 (pure f32 WMMA, no precision loss)
typedef __attribute__((ext_vector_type(2))) float v2f;
typedef __attribute__((ext_vector_type(8))) float v8f;

#define DIMS        128
#define NSEG        16
#define TILES       8              // DIMS / 16
#define PART_STRIDE 2064           // 16*128 sums + 16 counts
#define WAVES_PER_WG 8
#define THREADS     256
#define MAX_WG      1024

// Phase 1: each wave computes  C += A_onehot(16x4) * X_tile(4x16)  over its quads.
// A layout (32-bit A 16x4, ISA 7.12.2): lanes 0-15 -> {K=0 in v0, K=1 in v1},
//                                       lanes 16-31 -> {K=2 in v0, K=3 in v1}; M = lane%16.
// B layout (32-bit row-striped): v0: lanes 0-15 = row K=0, lanes 16-31 = row K=1;
//                                v1: lanes 0-15 = row K=2, lanes 16-31 = row K=3; N = lane%16.
// C/D layout: VGPR v, lanes 0-15 -> M=v, lanes 16-31 -> M=v+8; N = lane%16.
__global__ __launch_bounds__(THREADS)
void seg_pool_phase1(const float* __restrict__ x,
                     const int*   __restrict__ batch,
                     float*       __restrict__ partials,
                     int n, int numQuads, int totalWaves)
{
    __shared__ float lds[WAVES_PER_WG][NSEG * DIMS];  // 64 KB of the 320 KB WGP LDS
    __shared__ float ldsc[WAVES_PER_WG][32];

    const int tid   = threadIdx.x;
    const int lane  = tid & 31;
    const int wv    = tid >> 5;
    const int m     = lane & 15;     // segment row (A) / column (B) within tile
    const int khalf = lane >> 4;     // which K-half of the 4-row quad this lane holds

    // Wave-uniform by construction; readfirstlane moves it to an SGPR so the
    // whole q-loop runs on the SALU (s_cmp/s_cbranch, EXEC untouched -> WMMA-legal).
    const int gwave = __builtin_amdgcn_readfirstlane(blockIdx.x * WAVES_PER_WG + wv);
    const int last  = n - 1;

    v8f c[TILES];
#pragma unroll
    for (int t = 0; t < TILES; ++t) c[t] = (v8f)0.0f;
    float cnt = 0.0f;

    for (int q = gwave; q < numQuads; q += totalWaves) {
        const int base = q * 4;

        // ---- batch ids for the quad: uniform addresses -> scalar (SMEM) loads ----
        const int b0 = batch[min(base,     last)];
        const int b1 = batch[min(base + 1, last)];
        const int b2 = batch[min(base + 2, last)];
        const int b3 = batch[min(base + 3, last)];

        // ---- A fragment: one-hot of batch id vs this lane's segment ----
        const int  bs0 = khalf ? b2 : b0;
        const int  bs1 = khalf ? b3 : b1;
        const bool v0  = (base + 2 * khalf)     < n;
        const bool v1  = (base + 2 * khalf + 1) < n;
        const float a0 = (v0 && bs0 == m) ? 1.0f : 0.0f;
        const float a1 = (v1 && bs1 == m) ? 1.0f : 0.0f;
        cnt += a0 + a1;                      // segment counts come for free
        v2f a; a.x = a0; a.y = a1;

        // ---- B fragments: 4 rows x 16 cols per tile, clamped addresses (finite data) ----
        const int rb0 = min(base + khalf,     last);
        const int rb1 = min(base + 2 + khalf, last);
        const float* p0 = x + (long long)rb0 * DIMS + m;
        const float* p1 = x + (long long)rb1 * DIMS + m;
#pragma unroll
        for (int t = 0; t < TILES; ++t) {
            v2f b; b.x = p0[t * 16]; b.y = p1[t * 16];
            c[t] = __builtin_amdgcn_wmma_f32_16x16x4_f32(
                       false, a, false, b, (short)0, c[t], false, false);
        }
    }

    // ---- deterministic cross-wave merge through LDS (no atomics) ----
#pragma unroll
    for (int t = 0; t < TILES; ++t) {
#pragma unroll
        for (int v = 0; v < 8; ++v) {
            const int seg = (lane < 16) ? v : v + 8;
            lds[wv][seg * DIMS + (t * 16 + m)] = c[t][v];
        }
    }
    ldsc[wv][lane] = cnt;
    __syncthreads();

    float* dst = partials + (size_t)blockIdx.x * PART_STRIDE;
    for (int i = tid; i < NSEG * DIMS; i += THREADS) {
        float s = 0.0f;
#pragma unroll
        for (int w = 0; w < WAVES_PER_WG; ++w) s += lds[w][i];
        dst[i] = s;
    }
    if (tid < NSEG) {
        float s = 0.0f;
#pragma unroll
        for (int w = 0; w < WAVES_PER_WG; ++w) s += ldsc[w][tid] + ldsc[w][tid + 16];
        dst[NSEG * DIMS + tid] = s;
    }
}

// Phase 2: fold G partials, divide by max(count, 1).
__global__ __launch_bounds__(256)
void seg_pool_phase2(const float* __restrict__ partials,
                     float*       __restrict__ out, int G)
{
    const int idx = blockIdx.x * blockDim.x + threadIdx.x;
    if (idx >= NSEG * DIMS) return;
    const int j = idx >> 7;   // segment id (idx / 128)
    float s = 0.0f, cntv = 0.0f;
    for (int g = 0; g < G; ++g) {
        const float* p = partials + (size_t)g * PART_STRIDE;
        s    += p[idx];
        cntv += p[NSEG * DIMS + j];
    }
    out[idx] = s / fmaxf(cntv, 1.0f);
}

extern "C" void kernel_launch(void* const* d_in, const int* in_sizes, int n_in,
                              void* d_out, int out_size, void* d_ws, size_t ws_size,
                              hipStream_t stream)
{
    const float* x     = (const float*)d_in[0];
    const int*   batch = (const int*)d_in[1];
    // d_in[2] = num_segments scalar (fixed at 16 here)

    const int n = in_sizes[0] / DIMS;   // N rows
    float* partials = (float*)d_ws;

    int G = (int)(ws_size / (PART_STRIDE * sizeof(float)));
    if (G > MAX_WG) G = MAX_WG;
    if (G < 1) G = 1;

    const int numQuads   = (n + 3) / 4;
    const int totalWaves = G * WAVES_PER_WG;

    seg_pool_phase1<<<G, THREADS, 0, stream>>>(x, batch, partials, n, numQuads, totalWaves);
    seg_pool_phase2<<<(NSEG * DIMS + 255) / 256, 256, 0, stream>>>(partials, (float*)d_out, G);
}